// OrbBlock_65335042506809
// MI455X (gfx1250) — compile-verified
//
#include <hip/hip_runtime.h>
#include <math.h>

// ---------------------------------------------------------------------------
// OrbBlock for MI455X (gfx1250): fp32 WMMA (V_WMMA_F32_16X16X4_F32) GEMMs.
// N=20000 nodes, K=16 neighbours, D=128, H=256.
// Weights are pre-transposed per call so B-fragments are single b64 loads.
// ---------------------------------------------------------------------------

#define N_NODES 20000
#define KNBR    16
#define DIM     128
#define HID     256
#define IN3     384
#define MB      32            // rows per block (2 nodes * 16 neighbours)
#define LDX     388           // padded LDS strides (mod 64 != 0 -> no bank conflicts)
#define LDH     260
#define LDP     132
#define EPSV    1e-5f

typedef __attribute__((ext_vector_type(2))) float v2f;
typedef __attribute__((ext_vector_type(8))) float v8f;

__device__ __forceinline__ v8f wmma4(v2f a, v2f b, v8f c) {
  // D = A(16x4,f32) * B(4x16,f32) + C(16x16,f32)
  return __builtin_amdgcn_wmma_f32_16x16x4_f32(false, a, false, b, (short)0, c,
                                               false, false);
}

// ---- fast transcendentals on the CDNA5 TRANS pipe -------------------------
#if __has_builtin(__builtin_amdgcn_tanhf)
#define HW_TANH(x) __builtin_amdgcn_tanhf(x)
#elif __has_builtin(__builtin_amdgcn_tanh_f32)
#define HW_TANH(x) __builtin_amdgcn_tanh_f32(x)
#else
#define HW_TANH(x) tanhf(x)
#endif

__device__ __forceinline__ float exp_fast(float x) {
#if __has_builtin(__builtin_amdgcn_exp2f)
  return __builtin_amdgcn_exp2f(x * 1.4426950408889634f);  // v_exp_f32
#else
  return __expf(x);
#endif
}
__device__ __forceinline__ float rcp_fast(float x) {
#if __has_builtin(__builtin_amdgcn_rcpf)
  return __builtin_amdgcn_rcpf(x);                          // v_rcp_f32
#else
  return 1.0f / x;
#endif
}

__device__ __forceinline__ float gelu_tanh(float x) {
  const float inner = 0.7978845608028654f * (x + 0.044715f * x * x * x);
  return 0.5f * x * (1.0f + HW_TANH(inner));
}
__device__ __forceinline__ float sigmoidf(float x) {
  return rcp_fast(1.0f + exp_fast(-x));
}

// ---------------------------------------------------------------------------
// GEMM micro-kernels. A (16|32 rows x kdim) in LDS; W transposed: Wt[c*kdim+k].
// B-fragment = one contiguous v2f (global_load_b64) per matrix per k-step.
// ---------------------------------------------------------------------------
__device__ __forceinline__ void gemm_dual(
    const float* __restrict__ Xs, int ldx,
    const float* __restrict__ W0t, const float* __restrict__ W1t,
    int kdim, int col0, int lane,
    v8f& c00, v8f& c01, v8f& c10, v8f& c11)
{
  const int n = lane & 15, h = lane >> 4;
  const float* x0 = Xs + n * ldx + 2 * h;          // A: row=n, K = k + 2h (+1)
  const float* x1 = x0 + 16 * ldx;
  const float* w0 = W0t + (size_t)(col0 + n) * kdim + 2 * h;  // B: col=n
  const float* w1 = W1t + (size_t)(col0 + n) * kdim + 2 * h;
  #pragma unroll 4
  for (int k = 0; k < kdim; k += 4) {
    v2f a0 = *(const v2f*)(x0 + k);
    v2f a1 = *(const v2f*)(x1 + k);
    v2f b0 = *(const v2f*)(w0 + k);
    v2f b1 = *(const v2f*)(w1 + k);
    c00 = wmma4(a0, b0, c00);
    c01 = wmma4(a1, b0, c01);
    c10 = wmma4(a0, b1, c10);
    c11 = wmma4(a1, b1, c11);
  }
}

__device__ __forceinline__ void gemm_single(
    const float* __restrict__ Xs, int ldx,
    const float* __restrict__ Wt, int kdim, int col0, int lane,
    v8f& c0, v8f& c1)
{
  const int n = lane & 15, h = lane >> 4;
  const float* x0 = Xs + n * ldx + 2 * h;
  const float* x1 = x0 + 16 * ldx;
  const float* w  = Wt + (size_t)(col0 + n) * kdim + 2 * h;
  #pragma unroll 4
  for (int k = 0; k < kdim; k += 4) {
    v2f a0 = *(const v2f*)(x0 + k);
    v2f a1 = *(const v2f*)(x1 + k);
    v2f b  = *(const v2f*)(w + k);
    c0 = wmma4(a0, b, c0);
    c1 = wmma4(a1, b, c1);
  }
}

// ---------------------------------------------------------------------------
// Kernel 0a: zero the outgoing-accumulator workspace
// ---------------------------------------------------------------------------
__global__ void zero_f4(float4* __restrict__ p, int n4) {
  int i = blockIdx.x * blockDim.x + threadIdx.x;
  if (i < n4) p[i] = make_float4(0.f, 0.f, 0.f, 0.f);
}

// ---------------------------------------------------------------------------
// Kernel 0b: weight transpose  dst[c*kdim+k] = src[k*ncol+c]
// ---------------------------------------------------------------------------
__global__ __launch_bounds__(256)
void transpose_w(const float* __restrict__ src, float* __restrict__ dst,
                 int kdim, int ncol)
{
  __shared__ float tile[32][33];
  const int kb = blockIdx.x * 32, cb = blockIdx.y * 32;
  const int tx = threadIdx.x & 31, ty = threadIdx.x >> 5;   // 32 x 8
  #pragma unroll
  for (int i = ty; i < 32; i += 8)
    tile[i][tx] = src[(size_t)(kb + i) * ncol + cb + tx];
  __syncthreads();
  #pragma unroll
  for (int i = ty; i < 32; i += 8)
    dst[(size_t)(cb + i) * kdim + kb + tx] = tile[tx][i];
}

// ---------------------------------------------------------------------------
// Kernel 1: pair stage. One block = 2 nodes = 32 pair rows. 8 waves.
// ---------------------------------------------------------------------------
__global__ __launch_bounds__(256)
void pair_stage(const float* __restrict__ local,
                const float* __restrict__ pair,
                const int*   __restrict__ neighbours,
                const float* __restrict__ mask,
                const float* __restrict__ Wg1t, const float* __restrict__ bg1,
                const float* __restrict__ Wv1t, const float* __restrict__ bv1,
                const float* __restrict__ Wo1t, const float* __restrict__ bo1,
                const float* __restrict__ gp,   const float* __restrict__ bp,
                const float* __restrict__ Wint, const float* __restrict__ Woutt,
                float* __restrict__ pair_out,
                float* __restrict__ incoming,
                float* __restrict__ outgoing)
{
  __shared__ float Xs[MB * LDX];        // [pair | local_self | local_nbr]
  __shared__ float Hs[MB * LDH];        // hidden = gelu(g)*v
  __shared__ float PUs[MB * LDP];       // pair update (pre/post LN)
  __shared__ float red[MB][8], red2[MB][8];
  __shared__ float meanv[MB], rstdv[MB];
  __shared__ float pmaskS[MB];
  __shared__ int   nbrS[MB];

  const int tid  = threadIdx.x;
  const int lane = tid & 31;
  const int wave = tid >> 5;
  const size_t rowBase = (size_t)blockIdx.x * MB;  // flat pair-row base
  const int node0 = blockIdx.x * 2;

  // ---- stage 1: gather features into LDS -------------------------------
  {
    const int m = tid >> 3, s = tid & 7;          // 8 threads per row
    const int node = node0 + (m >> 4);
    const size_t grow = rowBase + m;
    const int nb = neighbours[grow];
    const int gi = nb < 0 ? nb + N_NODES : nb;    // JAX negative-index wrap
    if (s == 0) {
      nbrS[m]   = gi;
      pmaskS[m] = (nb != -1 && mask[gi] > 0.0f) ? 1.0f : 0.0f;
    }
    const float4* __restrict__ p4 = (const float4*)(pair  + grow * DIM);
    const float4* __restrict__ l4 = (const float4*)(local + (size_t)node * DIM);
    const float4* __restrict__ g4 = (const float4*)(local + (size_t)gi   * DIM);
    float4* __restrict__ xa = (float4*)(Xs + m * LDX);
    float4* __restrict__ xb = (float4*)(Xs + m * LDX + DIM);
    float4* __restrict__ xc = (float4*)(Xs + m * LDX + 2 * DIM);
    #pragma unroll
    for (int j = 0; j < 4; ++j) {
      const int i = s + j * 8;
      xa[i] = p4[i];
      xb[i] = l4[i];
      xc[i] = g4[i];
    }
  }
  __syncthreads();

  // ---- stage 2: gate & value GEMMs (384 -> 256), gelu(g)*v -> Hs -------
  #pragma unroll
  for (int t = 0; t < 2; ++t) {
    const int col0 = (wave + t * 8) * 16;
    v8f g0 = {}, g1 = {}, v0 = {}, v1 = {};
    gemm_dual(Xs, LDX, Wg1t, Wv1t, IN3, col0, lane, g0, g1, v0, v1);
    const int n = lane & 15, h = lane >> 4;
    const float bgv = bg1[col0 + n];
    const float bvv = bv1[col0 + n];
    float* h0 = Hs + (8 * h) * LDH + col0 + n;
    float* h1 = Hs + (16 + 8 * h) * LDH + col0 + n;
    #pragma unroll
    for (int r = 0; r < 8; ++r) {
      h0[r * LDH] = gelu_tanh(g0[r] + bgv) * (v0[r] + bvv);
      h1[r * LDH] = gelu_tanh(g1[r] + bgv) * (v1[r] + bvv);
    }
  }
  __syncthreads();

  // ---- stage 3: output projection (256 -> 128) -> PUs ------------------
  {
    const int col0 = wave * 16;
    v8f o0 = {}, o1 = {};
    gemm_single(Hs, LDH, Wo1t, HID, col0, lane, o0, o1);
    const int n = lane & 15, h = lane >> 4;
    const float bov = bo1[col0 + n];
    float* p0 = PUs + (8 * h) * LDP + col0 + n;
    float* p1 = PUs + (16 + 8 * h) * LDP + col0 + n;
    #pragma unroll
    for (int r = 0; r < 8; ++r) {
      p0[r * LDP] = o0[r] + bov;
      p1[r * LDP] = o1[r] + bov;
    }
  }
  __syncthreads();

  // ---- stage 4: LayerNorm + pair residual ------------------------------
  {
    const int m = tid >> 3, s = tid & 7;
    float sum = 0.f, sq = 0.f;
    const float* pr = PUs + m * LDP + s * 16;
    #pragma unroll
    for (int c = 0; c < 16; ++c) { float v = pr[c]; sum += v; sq += v * v; }
    red[m][s] = sum; red2[m][s] = sq;
    __syncthreads();
    if (s == 0) {
      float S = 0.f, Q = 0.f;
      #pragma unroll
      for (int i = 0; i < 8; ++i) { S += red[m][i]; Q += red2[m][i]; }
      const float mu  = S * (1.0f / DIM);
      const float var = Q * (1.0f / DIM) - mu * mu;
      meanv[m] = mu;
      rstdv[m] = rsqrtf(var + EPSV);
    }
    __syncthreads();
    const float mu = meanv[m], rs = rstdv[m];
    float* pw = PUs + m * LDP;
    const size_t grow = (rowBase + m) * DIM;
    for (int c = s * 16; c < s * 16 + 16; ++c) {
      const float u = (pw[c] - mu) * rs * gp[c] + bp[c];
      pw[c] = u;                                // normalized update, reused below
      pair_out[grow + c] = pair[grow + c] + u;  // residual output
    }
  }
  __syncthreads();

  // ---- stage 5: sigmoid(pair@Win/Wout) gates, incoming sum, outgoing scatter
  {
    const int col0 = wave * 16;                 // wave owns 16 of the 128 cols
    v8f i0 = {}, i1 = {}, o0 = {}, o1 = {};
    gemm_dual(Xs, LDX, Wint, Woutt, DIM, col0, lane, i0, i1, o0, o1);
    const int n = lane & 15, h = lane >> 4;
    #pragma unroll
    for (int rt = 0; rt < 2; ++rt) {            // row-tile == one node's 16 rows
      const v8f ci = rt ? i1 : i0;
      const v8f co = rt ? o1 : o0;
      float partial = 0.f;
      #pragma unroll
      for (int r = 0; r < 8; ++r) {
        const int row = rt * 16 + r + 8 * h;
        const float pu = PUs[row * LDP + col0 + n];
        const float pm = pmaskS[row];
        partial += sigmoidf(ci[r]) * pu * pm;
        if (pm > 0.f) {
          const float ov = sigmoidf(co[r]) * pu;
          atomicAdd(&outgoing[(size_t)nbrS[row] * DIM + col0 + n], ov);
        }
      }
      // lanes L and L+16 hold the two 8-row halves of the same column
      partial += __shfl_xor(partial, 16, 32);
      if (lane < 16)
        incoming[(size_t)(node0 + rt) * DIM + col0 + n] = partial;
    }
  }
}

// ---------------------------------------------------------------------------
// Kernel 2: local stage. One block = 32 nodes. 8 waves.
// ---------------------------------------------------------------------------
__global__ __launch_bounds__(256)
void local_stage(const float* __restrict__ local,
                 const float* __restrict__ incoming,
                 const float* __restrict__ outgoing,
                 const float* __restrict__ Wg2t, const float* __restrict__ bg2,
                 const float* __restrict__ Wv2t, const float* __restrict__ bv2,
                 const float* __restrict__ Wo2t, const float* __restrict__ bo2,
                 const float* __restrict__ gl,   const float* __restrict__ bl,
                 float* __restrict__ local_out)
{
  __shared__ float Xs[MB * LDX];
  __shared__ float Hs[MB * LDH];
  __shared__ float PUs[MB * LDP];
  __shared__ float red[MB][8], red2[MB][8];
  __shared__ float meanv[MB], rstdv[MB];

  const int tid  = threadIdx.x;
  const int lane = tid & 31;
  const int wave = tid >> 5;
  const size_t nodeBase = (size_t)blockIdx.x * MB;

  // ---- load [local | incoming | outgoing] ------------------------------
  {
    const int m = tid >> 3, s = tid & 7;
    const size_t node = nodeBase + m;
    const float4* __restrict__ l4 = (const float4*)(local    + node * DIM);
    const float4* __restrict__ i4 = (const float4*)(incoming + node * DIM);
    const float4* __restrict__ o4 = (const float4*)(outgoing + node * DIM);
    float4* __restrict__ xa = (float4*)(Xs + m * LDX);
    float4* __restrict__ xb = (float4*)(Xs + m * LDX + DIM);
    float4* __restrict__ xc = (float4*)(Xs + m * LDX + 2 * DIM);
    #pragma unroll
    for (int j = 0; j < 4; ++j) {
      const int i = s + j * 8;
      xa[i] = l4[i];
      xb[i] = i4[i];
      xc[i] = o4[i];
    }
  }
  __syncthreads();

  // ---- gate & value GEMMs ---------------------------------------------
  #pragma unroll
  for (int t = 0; t < 2; ++t) {
    const int col0 = (wave + t * 8) * 16;
    v8f g0 = {}, g1 = {}, v0 = {}, v1 = {};
    gemm_dual(Xs, LDX, Wg2t, Wv2t, IN3, col0, lane, g0, g1, v0, v1);
    const int n = lane & 15, h = lane >> 4;
    const float bgv = bg2[col0 + n];
    const float bvv = bv2[col0 + n];
    float* h0 = Hs + (8 * h) * LDH + col0 + n;
    float* h1 = Hs + (16 + 8 * h) * LDH + col0 + n;
    #pragma unroll
    for (int r = 0; r < 8; ++r) {
      h0[r * LDH] = gelu_tanh(g0[r] + bgv) * (v0[r] + bvv);
      h1[r * LDH] = gelu_tanh(g1[r] + bgv) * (v1[r] + bvv);
    }
  }
  __syncthreads();

  // ---- output projection ----------------------------------------------
  {
    const int col0 = wave * 16;
    v8f o0 = {}, o1 = {};
    gemm_single(Hs, LDH, Wo2t, HID, col0, lane, o0, o1);
    const int n = lane & 15, h = lane >> 4;
    const float bov = bo2[col0 + n];
    float* p0 = PUs + (8 * h) * LDP + col0 + n;
    float* p1 = PUs + (16 + 8 * h) * LDP + col0 + n;
    #pragma unroll
    for (int r = 0; r < 8; ++r) {
      p0[r * LDP] = o0[r] + bov;
      p1[r * LDP] = o1[r] + bov;
    }
  }
  __syncthreads();

  // ---- LayerNorm + residual -------------------------------------------
  {
    const int m = tid >> 3, s = tid & 7;
    float sum = 0.f, sq = 0.f;
    const float* pr = PUs + m * LDP + s * 16;
    #pragma unroll
    for (int c = 0; c < 16; ++c) { float v = pr[c]; sum += v; sq += v * v; }
    red[m][s] = sum; red2[m][s] = sq;
    __syncthreads();
    if (s == 0) {
      float S = 0.f, Q = 0.f;
      #pragma unroll
      for (int i = 0; i < 8; ++i) { S += red[m][i]; Q += red2[m][i]; }
      const float mu  = S * (1.0f / DIM);
      const float var = Q * (1.0f / DIM) - mu * mu;
      meanv[m] = mu;
      rstdv[m] = rsqrtf(var + EPSV);
    }
    __syncthreads();
    const float mu = meanv[m], rs = rstdv[m];
    const float* pw = PUs + m * LDP;
    const size_t grow = (nodeBase + m) * DIM;
    for (int c = s * 16; c < s * 16 + 16; ++c) {
      const float u = (pw[c] - mu) * rs * gl[c] + bl[c];
      local_out[grow + c] = local[grow + c] + u;
    }
  }
}

// ---------------------------------------------------------------------------
extern "C" void kernel_launch(void* const* d_in, const int* in_sizes, int n_in,
                              void* d_out, int out_size, void* d_ws, size_t ws_size,
                              hipStream_t stream)
{
  const float* local = (const float*)d_in[0];
  const float* pair  = (const float*)d_in[1];
  const int*   nbr   = (const int*)  d_in[2];
  const float* mask  = (const float*)d_in[3];
  const float* Wg1 = (const float*)d_in[4];  const float* bg1 = (const float*)d_in[5];
  const float* Wv1 = (const float*)d_in[6];  const float* bv1 = (const float*)d_in[7];
  const float* Wo1 = (const float*)d_in[8];  const float* bo1 = (const float*)d_in[9];
  const float* gp  = (const float*)d_in[10]; const float* bp  = (const float*)d_in[11];
  const float* Win = (const float*)d_in[12]; const float* Wout= (const float*)d_in[13];
  const float* Wg2 = (const float*)d_in[14]; const float* bg2 = (const float*)d_in[15];
  const float* Wv2 = (const float*)d_in[16]; const float* bv2 = (const float*)d_in[17];
  const float* Wo2 = (const float*)d_in[18]; const float* bo2 = (const float*)d_in[19];
  const float* gl  = (const float*)d_in[20]; const float* bl  = (const float*)d_in[21];

  float* local_out = (float*)d_out;                         // [N, D]
  float* pair_out  = local_out + (size_t)N_NODES * DIM;     // [N, K, D]

  // workspace layout (floats)
  float* ws = (float*)d_ws;
  float* incoming = ws;                                     // [N, D]
  float* outgoing = incoming + (size_t)N_NODES * DIM;       // [N, D]
  float* wt       = outgoing + (size_t)N_NODES * DIM;       // transposed weights
  float* Wg1t  = wt;               // 256*384
  float* Wv1t  = Wg1t  + HID * IN3;
  float* Wo1t  = Wv1t  + HID * IN3;   // 128*256
  float* Wint  = Wo1t  + DIM * HID;   // 128*128
  float* Woutt = Wint  + DIM * DIM;
  float* Wg2t  = Woutt + DIM * DIM;
  float* Wv2t  = Wg2t  + HID * IN3;
  float* Wo2t  = Wv2t  + HID * IN3;

  // ---- transpose all weight matrices (tiny: ~2MB, L2-resident) ---------
  {
    dim3 gA(IN3 / 32, HID / 32);  // 384 x 256
    transpose_w<<<gA, 256, 0, stream>>>(Wg1, Wg1t, IN3, HID);
    transpose_w<<<gA, 256, 0, stream>>>(Wv1, Wv1t, IN3, HID);
    transpose_w<<<gA, 256, 0, stream>>>(Wg2, Wg2t, IN3, HID);
    transpose_w<<<gA, 256, 0, stream>>>(Wv2, Wv2t, IN3, HID);
    dim3 gO(HID / 32, DIM / 32);  // 256 x 128
    transpose_w<<<gO, 256, 0, stream>>>(Wo1, Wo1t, HID, DIM);
    transpose_w<<<gO, 256, 0, stream>>>(Wo2, Wo2t, HID, DIM);
    dim3 gG(DIM / 32, DIM / 32);  // 128 x 128
    transpose_w<<<gG, 256, 0, stream>>>(Win,  Wint,  DIM, DIM);
    transpose_w<<<gG, 256, 0, stream>>>(Wout, Woutt, DIM, DIM);
  }

  // zero the scatter-add accumulator (incoming is plain-stored, no init needed)
  const int n4 = N_NODES * DIM / 4;
  zero_f4<<<(n4 + 255) / 256, 256, 0, stream>>>((float4*)outgoing, n4);

  pair_stage<<<N_NODES / 2, 256, 0, stream>>>(
      local, pair, nbr, mask,
      Wg1t, bg1, Wv1t, bv1, Wo1t, bo1, gp, bp, Wint, Woutt,
      pair_out, incoming, outgoing);

  local_stage<<<N_NODES / MB, 256, 0, stream>>>(
      local, incoming, outgoing,
      Wg2t, bg2, Wv2t, bv2, Wo2t, bo2, gl, bl,
      local_out);
}